// kGNN_72541997629470
// MI455X (gfx1250) — compile-verified
//
#include <hip/hip_runtime.h>

#define HDIM 256           // feature / hidden dim (D == H == 256)

typedef float v2f __attribute__((ext_vector_type(2)));
typedef float v8f __attribute__((ext_vector_type(8)));

// ---------------------------------------------------------------------------
// Zero a float buffer (agg accumulator must start at 0 each layer).
// ---------------------------------------------------------------------------
__global__ void kZeroF32(float* __restrict__ p, int n) {
  int i = blockIdx.x * blockDim.x + threadIdx.x;
  if (i < n) p[i] = 0.0f;
}

// ---------------------------------------------------------------------------
// Edge scatter-add: agg[dst[e], :] += x[src[e], :]
// One thread per (edge, 4-feature quad): float4 load from the source row,
// 4 hardware global_atomic_add_f32 into the destination row. Both the source
// rows and the 51.2MB agg buffer live in the 192MB L2, so these atomics are
// L2-resident and coalesced within an edge.
// ---------------------------------------------------------------------------
__global__ void kScatterAdd(const float* __restrict__ x,
                            const int* __restrict__ src,
                            const int* __restrict__ dst,
                            float* __restrict__ agg,
                            int nEdges) {
  long long t = (long long)blockIdx.x * blockDim.x + threadIdx.x;
  long long total = (long long)nEdges * (HDIM / 4);
  if (t >= total) return;
  int e = (int)(t >> 6);        // HDIM/4 == 64 quads per edge
  int q = (int)(t & 63);
  int s = src[e];
  int d = dst[e];
  float4 v = *(const float4*)(x + (long long)s * HDIM + q * 4);
  float* o = agg + (long long)d * HDIM + q * 4;
  unsafeAtomicAdd(o + 0, v.x);   // global_atomic_add_f32 (no CAS loop)
  unsafeAtomicAdd(o + 1, v.y);
  unsafeAtomicAdd(o + 2, v.z);
  unsafeAtomicAdd(o + 3, v.w);
}

// ---------------------------------------------------------------------------
// Fused GraphConv GEMM:  out = ELU( agg @ Wrel + x @ Wroot + b )
//
// Block = 256 threads = 8 wave32s. blockIdx.y picks a 16-column slab; the
// block stages Wrel[:,c0:c0+16] and Wroot[:,c0:c0+16] (16KB each) into LDS.
// Each wave owns one 16x16 output tile (8 consecutive row-tiles per block)
// and accumulates K=256 twice with V_WMMA_F32_16X16X4_F32 (fp32-exact).
//
// A fragment (16x4 f32, ISA layout): lanes 0-15 -> {A[m][k],A[m][k+1]},
//   lanes 16-31 -> {A[m][k+2],A[m][k+3]}  -> one float2 load per lane.
// B fragment (4x16): lanes 0-15 -> {B[k][n],B[k+1][n]},
//   lanes 16-31 -> {B[k+2][n],B[k+3][n]}  -> two ds_load_b32 per lane.
// C/D (16x16 f32): vgpr i, lanes 0-15 -> row i, lanes 16-31 -> row i+8.
// ---------------------------------------------------------------------------
__global__ __launch_bounds__(256)
void kGraphConvGemm(const float* __restrict__ agg,
                    const float* __restrict__ xin,
                    const float* __restrict__ Wrel,
                    const float* __restrict__ Wroot,
                    const float* __restrict__ bias,
                    float* __restrict__ out,
                    int nNodes) {
  __shared__ float sRel[HDIM * 16];
  __shared__ float sRoot[HDIM * 16];

  const int col0 = blockIdx.y * 16;

  // Stage both 256x16 weight slabs into LDS (float4 granularity).
  for (int i = threadIdx.x; i < HDIM * 4; i += blockDim.x) {
    int k = i >> 2, n4 = i & 3;
    ((float4*)sRel)[i]  = *(const float4*)(Wrel  + (long long)k * HDIM + col0 + n4 * 4);
    ((float4*)sRoot)[i] = *(const float4*)(Wroot + (long long)k * HDIM + col0 + n4 * 4);
  }
  __syncthreads();

  const int wave  = threadIdx.x >> 5;
  const int lane  = threadIdx.x & 31;
  const int row0  = (blockIdx.x * 8 + wave) * 16;
  if (row0 >= nNodes) return;            // N=50000 is a multiple of 16: full tiles only

  const int lhalf = lane >> 4;           // 0: K pair {k,k+1}, 1: K pair {k+2,k+3}
  const int l16   = lane & 15;

  v8f c = {};                             // fp32 accumulator, starts at 0

  // Pass 1: agg @ Wrel
  {
    const float* arow = agg + (long long)(row0 + l16) * HDIM + lhalf * 2;
    #pragma unroll 4
    for (int kk = 0; kk < HDIM; kk += 4) {
      v2f a = *(const v2f*)(arow + kk);
      v2f b;
      b.x = sRel[(kk + lhalf * 2 + 0) * 16 + l16];
      b.y = sRel[(kk + lhalf * 2 + 1) * 16 + l16];
      c = __builtin_amdgcn_wmma_f32_16x16x4_f32(false, a, false, b,
                                                (short)0, c, false, false);
    }
  }
  // Pass 2: + x @ Wroot (same accumulator)
  {
    const float* xrow = xin + (long long)(row0 + l16) * HDIM + lhalf * 2;
    #pragma unroll 4
    for (int kk = 0; kk < HDIM; kk += 4) {
      v2f a = *(const v2f*)(xrow + kk);
      v2f b;
      b.x = sRoot[(kk + lhalf * 2 + 0) * 16 + l16];
      b.y = sRoot[(kk + lhalf * 2 + 1) * 16 + l16];
      c = __builtin_amdgcn_wmma_f32_16x16x4_f32(false, a, false, b,
                                                (short)0, c, false, false);
    }
  }

  // Bias + ELU + store (C/D VGPR layout).
  const float bv = bias[col0 + l16];
  #pragma unroll
  for (int i = 0; i < 8; ++i) {
    int m = i + lhalf * 8;
    float v = c[i] + bv;
    v = v > 0.0f ? v : (__expf(v) - 1.0f);   // jax.nn.elu, alpha=1
    out[(long long)(row0 + m) * HDIM + col0 + l16] = v;
  }
}

// ---------------------------------------------------------------------------
// Driver: 3 layers of { zero agg ; scatter-add ; fused WMMA GEMM + ELU }.
// Workspace: agg | h1 | h2, each N*H floats (3 * 51.2 MB).
// ---------------------------------------------------------------------------
extern "C" void kernel_launch(void* const* d_in, const int* in_sizes, int n_in,
                              void* d_out, int out_size, void* d_ws, size_t ws_size,
                              hipStream_t stream) {
  const float* x     = (const float*)d_in[0];
  const int*   eidx  = (const int*)  d_in[1];
  const float* W1r   = (const float*)d_in[2];
  const float* b1    = (const float*)d_in[3];
  const float* W1o   = (const float*)d_in[4];
  const float* W2r   = (const float*)d_in[5];
  const float* b2    = (const float*)d_in[6];
  const float* W2o   = (const float*)d_in[7];
  const float* W3r   = (const float*)d_in[8];
  const float* b3    = (const float*)d_in[9];
  const float* W3o   = (const float*)d_in[10];

  const int nNodes = in_sizes[0] / HDIM;   // 50000
  const int nEdges = in_sizes[1] / 2;      // 800000
  const int*   src = eidx;                 // edge_index[0,:]
  const int*   dst = eidx + nEdges;        // edge_index[1,:]

  const long long NH = (long long)nNodes * HDIM;
  float* agg = (float*)d_ws;
  float* h1  = agg + NH;
  float* h2  = h1 + NH;
  float* outp = (float*)d_out;

  const int zblocks = (int)((NH + 255) / 256);
  const long long sitems = (long long)nEdges * (HDIM / 4);
  const int sblocks = (int)((sitems + 255) / 256);
  dim3 ggrid((nNodes / 16 + 7) / 8, HDIM / 16);   // 8 row-tiles/block x 16 col slabs

  // Layer 1: h1 = ELU(agg(x) @ W1r + b1 + x @ W1o)
  kZeroF32<<<zblocks, 256, 0, stream>>>(agg, (int)NH);
  kScatterAdd<<<sblocks, 256, 0, stream>>>(x, src, dst, agg, nEdges);
  kGraphConvGemm<<<ggrid, 256, 0, stream>>>(agg, x, W1r, W1o, b1, h1, nNodes);

  // Layer 2
  kZeroF32<<<zblocks, 256, 0, stream>>>(agg, (int)NH);
  kScatterAdd<<<sblocks, 256, 0, stream>>>(h1, src, dst, agg, nEdges);
  kGraphConvGemm<<<ggrid, 256, 0, stream>>>(agg, h1, W2r, W2o, b2, h2, nNodes);

  // Layer 3 -> d_out
  kZeroF32<<<zblocks, 256, 0, stream>>>(agg, (int)NH);
  kScatterAdd<<<sblocks, 256, 0, stream>>>(h2, src, dst, agg, nEdges);
  kGraphConvGemm<<<ggrid, 256, 0, stream>>>(agg, h2, W3r, W3o, b3, outp, nNodes);
}